// WaveletLayer_80942953660692
// MI455X (gfx1250) — compile-verified
//
#include <hip/hip_runtime.h>

// Fused: out = diag_s * DWT3( diag_g * P( DWT3( diag_b * x ) ) ), Haar 2-tap,
// D = 4096, rows independent. One block (256 threads, 8 wave32) per row.
// Row staged in LDS via gfx1250 async global->LDS copies (ASYNCcnt path).

#define DD 4096
#define NT 256

struct W8 { float l3, h3, h20, h21, h10, h11, h12, h13; };

// 3-level 2-tap DWT on 8 contiguous elements. Filter taps carry the 1/sqrt2
// scaling per level, so outputs are exactly the banded coefficients.
__device__ __forceinline__ W8 dwt8(const float* a,
                                   float fl0, float fl1, float fh0, float fh1) {
  W8 r;
  float l10 = fl0 * a[0] + fl1 * a[1];
  float l11 = fl0 * a[2] + fl1 * a[3];
  float l12 = fl0 * a[4] + fl1 * a[5];
  float l13 = fl0 * a[6] + fl1 * a[7];
  r.h10 = fh0 * a[0] + fh1 * a[1];
  r.h11 = fh0 * a[2] + fh1 * a[3];
  r.h12 = fh0 * a[4] + fh1 * a[5];
  r.h13 = fh0 * a[6] + fh1 * a[7];
  float l20 = fl0 * l10 + fl1 * l11;
  float l21 = fl0 * l12 + fl1 * l13;
  r.h20 = fh0 * l10 + fh1 * l11;
  r.h21 = fh0 * l12 + fh1 * l13;
  r.l3  = fl0 * l20 + fl1 * l21;
  r.h3  = fh0 * l20 + fh1 * l21;
  return r;
}

__global__ __launch_bounds__(NT) void wavelet_fused_kernel(
    const float* __restrict__ x,
    const float* __restrict__ diag_s,
    const float* __restrict__ diag_g,
    const float* __restrict__ diag_b,
    const float* __restrict__ dec_lo,
    const float* __restrict__ dec_hi,
    const int*   __restrict__ perm,
    float* __restrict__ out)
{
  __shared__ __align__(16) float xs[DD];   // staged input row
  __shared__ __align__(16) float mid[DD];  // after first DWT (pre-permutation)

  const int t = threadIdx.x;                       // 0..255, owns elems [16t,16t+16)
  const long long row = blockIdx.x;
  const float* xrow = x + row * (long long)DD;
  float*       orow = out + row * (long long)DD;

  const float fl0 = dec_lo[0], fl1 = dec_lo[1];    // uniform -> scalar loads
  const float fh0 = dec_hi[0], fh1 = dec_hi[1];

  // ---- Async stage: copy this thread's 16B x 4 into LDS (gfx1250 path) ----
  #pragma unroll
  for (int j = 0; j < 4; ++j) {
    const int e = 16 * t + 4 * j;
    unsigned lds_addr = (unsigned)(unsigned long long)(&xs[e]);
    unsigned long long gaddr = (unsigned long long)(xrow + e);
    asm volatile("global_load_async_to_lds_b128 %0, %1, off"
                 :: "v"(lds_addr), "v"(gaddr) : "memory");
  }
#if __has_builtin(__builtin_amdgcn_s_wait_asynccnt)
  __builtin_amdgcn_s_wait_asynccnt(0);
#else
  asm volatile("s_wait_asynccnt 0" ::: "memory");
#endif
  // Each thread reads only the region it loaded itself -> no barrier needed yet.

  // ---- Stage 1: a = x * diag_b, then 3-level DWT of two 8-groups ----
  float a[16];
  #pragma unroll
  for (int j = 0; j < 4; ++j) {
    float4 xv = *(const float4*)&xs[16 * t + 4 * j];
    float4 bv = *(const float4*)&diag_b[16 * t + 4 * j];
    a[4 * j + 0] = xv.x * bv.x;
    a[4 * j + 1] = xv.y * bv.y;
    a[4 * j + 2] = xv.z * bv.z;
    a[4 * j + 3] = xv.w * bv.w;
  }
  W8 r0 = dwt8(&a[0], fl0, fl1, fh0, fh1);   // group g0 = 2t
  W8 r1 = dwt8(&a[8], fl0, fl1, fh0, fh1);   // group g1 = 2t+1

  // Banded, vectorized LDS writes (offsets are 8/16B aligned by construction).
  { float2 o; o.x = r0.l3;  o.y = r1.l3;  *(float2*)&mid[2 * t]        = o; }
  { float2 o; o.x = r0.h3;  o.y = r1.h3;  *(float2*)&mid[512 + 2 * t]  = o; }
  { float4 o; o.x = r0.h20; o.y = r0.h21; o.z = r1.h20; o.w = r1.h21;
    *(float4*)&mid[1024 + 4 * t] = o; }
  { float4 o; o.x = r0.h10; o.y = r0.h11; o.z = r0.h12; o.w = r0.h13;
    *(float4*)&mid[2048 + 8 * t] = o; }
  { float4 o; o.x = r1.h10; o.y = r1.h11; o.z = r1.h12; o.w = r1.h13;
    *(float4*)&mid[2048 + 8 * t + 4] = o; }

  __syncthreads();

  // ---- Stage 2: permutation gather from LDS, * diag_g, second DWT ----
  float v[16];
  #pragma unroll
  for (int j = 0; j < 4; ++j) {
    int4   p  = *(const int4*)&perm[16 * t + 4 * j];
    float4 gv = *(const float4*)&diag_g[16 * t + 4 * j];
    v[4 * j + 0] = mid[p.x] * gv.x;
    v[4 * j + 1] = mid[p.y] * gv.y;
    v[4 * j + 2] = mid[p.z] * gv.z;
    v[4 * j + 3] = mid[p.w] * gv.w;
  }
  W8 q0 = dwt8(&v[0], fl0, fl1, fh0, fh1);
  W8 q1 = dwt8(&v[8], fl0, fl1, fh0, fh1);

  // ---- * diag_s at destination, banded coalesced stores to the output row ----
  float2 sl3 = *(const float2*)&diag_s[2 * t];
  float2 sh3 = *(const float2*)&diag_s[512 + 2 * t];
  float4 sh2 = *(const float4*)&diag_s[1024 + 4 * t];
  float4 s1a = *(const float4*)&diag_s[2048 + 8 * t];
  float4 s1b = *(const float4*)&diag_s[2048 + 8 * t + 4];

  { float2 o; o.x = q0.l3 * sl3.x;  o.y = q1.l3 * sl3.y;
    *(float2*)&orow[2 * t] = o; }
  { float2 o; o.x = q0.h3 * sh3.x;  o.y = q1.h3 * sh3.y;
    *(float2*)&orow[512 + 2 * t] = o; }
  { float4 o; o.x = q0.h20 * sh2.x; o.y = q0.h21 * sh2.y;
    o.z = q1.h20 * sh2.z; o.w = q1.h21 * sh2.w;
    *(float4*)&orow[1024 + 4 * t] = o; }
  { float4 o; o.x = q0.h10 * s1a.x; o.y = q0.h11 * s1a.y;
    o.z = q0.h12 * s1a.z; o.w = q0.h13 * s1a.w;
    *(float4*)&orow[2048 + 8 * t] = o; }
  { float4 o; o.x = q1.h10 * s1b.x; o.y = q1.h11 * s1b.y;
    o.z = q1.h12 * s1b.z; o.w = q1.h13 * s1b.w;
    *(float4*)&orow[2048 + 8 * t + 4] = o; }
}

extern "C" void kernel_launch(void* const* d_in, const int* in_sizes, int n_in,
                              void* d_out, int out_size, void* d_ws, size_t ws_size,
                              hipStream_t stream) {
  const float* x      = (const float*)d_in[0];
  const float* diag_s = (const float*)d_in[1];
  const float* diag_g = (const float*)d_in[2];
  const float* diag_b = (const float*)d_in[3];
  const float* dec_lo = (const float*)d_in[4];
  const float* dec_hi = (const float*)d_in[5];
  const int*   perm   = (const int*)d_in[6];
  // d_in[7] = scales (==3, structure hardcoded); dec filters are 2-tap Haar.
  float* out = (float*)d_out;

  const int D    = in_sizes[1];          // 4096
  const int rows = in_sizes[0] / D;      // 4096
  (void)n_in; (void)out_size; (void)d_ws; (void)ws_size; (void)D;

  wavelet_fused_kernel<<<rows, NT, 0, stream>>>(
      x, diag_s, diag_g, diag_b, dec_lo, dec_hi, perm, out);
}